// ToDenseBEVConvolution_58205396796001
// MI455X (gfx1250) — compile-verified
//
#include <hip/hip_runtime.h>

typedef __attribute__((ext_vector_type(2))) float v2f;
typedef __attribute__((ext_vector_type(8))) float v8f;

#define C_IN   64
#define C_OUT  128
#define NKER   16
#define BEV_H  512
#define BEV_W  512
#define HW     (BEV_H * BEV_W)

// ---------------- workspace layout (int units) ----------------
// ws[0..15]   : bucket counts
// ws[16..31]  : bucket offsets
// ws[32..47]  : bucket cursors
// ws[48]      : numTiles
// ws[64..]    : tiles, 3 ints each {kidx, start, cnt}
// ws[64+3*maxTiles ..] : perm[N]

__global__ void zero_meta_k(int* ws) {
    int t = threadIdx.x;
    if (t < 16) ws[t] = 0;          // counts
    if (t == 16) ws[48] = 0;        // numTiles
}

__global__ void count_k(const int* __restrict__ coords,
                        const int* __restrict__ stridePtr,
                        int* __restrict__ counts, int n) {
    int i = blockIdx.x * blockDim.x + threadIdx.x;
    if (i < n) {
        int h = (coords[4 * i + 1] / (*stridePtr)) & (NKER - 1);
        atomicAdd(&counts[h], 1);
    }
}

__global__ void scan_build_k(int* ws) {
    // single thread: 16 buckets, builds tile list
    int* counts   = ws;
    int* offsets  = ws + 16;
    int* cursor   = ws + 32;
    int* numTiles = ws + 48;
    int* tiles    = ws + 64;
    int off = 0, nt = 0;
    for (int k = 0; k < NKER; ++k) {
        offsets[k] = off;
        cursor[k]  = 0;
        int c = counts[k];
        for (int t = 0; t < c; t += 16) {
            tiles[3 * nt + 0] = k;
            tiles[3 * nt + 1] = off + t;
            tiles[3 * nt + 2] = (c - t) < 16 ? (c - t) : 16;
            ++nt;
        }
        off += c;
    }
    *numTiles = nt;
}

__global__ void scatter_k(const int* __restrict__ coords,
                          const int* __restrict__ stridePtr,
                          const int* __restrict__ offsets,
                          int* __restrict__ cursor,
                          int* __restrict__ perm, int n) {
    int i = blockIdx.x * blockDim.x + threadIdx.x;
    if (i < n) {
        int h = (coords[4 * i + 1] / (*stridePtr)) & (NKER - 1);
        int pos = offsets[h] + atomicAdd(&cursor[h], 1);
        perm[pos] = i;
    }
}

__global__ void zero_out_k(float4* __restrict__ out, int n4) {
    int i = blockIdx.x * blockDim.x + threadIdx.x;
    if (i < n4) out[i] = make_float4(0.f, 0.f, 0.f, 0.f);
}

// One block (8 waves) per tile of <=16 points, all sharing kernel index k.
// Wave w computes the 16x16 C tile for output channels [16w, 16w+16) with
// sixteen v_wmma_f32_16x16x4_f32 steps over K=64, then scatter-adds.
__global__ void __launch_bounds__(256)
gemm_scatter_k(const float* __restrict__ feats,
               const float* __restrict__ kern,     // [16][64][128]
               const int*   __restrict__ coords,   // [N][4]
               const int*   __restrict__ perm,
               const int*   __restrict__ tiles,
               const int*   __restrict__ numTilesPtr,
               const int*   __restrict__ stridePtr,
               float*       __restrict__ out) {
    int t = blockIdx.x;
    if (t >= *numTilesPtr) return;              // block-uniform exit

    __shared__ float A[16][C_IN + 4];           // +4 pad: conflict-free column reads
    __shared__ int   base[16];

    int k     = tiles[3 * t + 0];
    int start = tiles[3 * t + 1];
    int cnt   = tiles[3 * t + 2];
    int sv    = *stridePtr;

    int tid = threadIdx.x;
    // ---- stage A tile (16 rows x 64 f32), zero-padded ----
    int row  = tid >> 4;                        // 0..15
    int col4 = (tid & 15) * 4;                  // 0..60
    float4 v = make_float4(0.f, 0.f, 0.f, 0.f);
    if (row < cnt) {
        int p = perm[start + row];
        v = *(const float4*)(feats + p * C_IN + col4);
    }
    *(float4*)&A[row][col4] = v;
    // ---- per-row flat BEV base index ----
    if (tid < 16) {
        int b = -1;
        if (tid < cnt) {
            int p  = perm[start + tid];
            int4 c = *(const int4*)(coords + p * 4);
            int xb = c.x / sv, zb = c.z / sv, bb = c.w;
            b = bb * (C_OUT * HW) + xb * BEV_W + zb;
        }
        base[tid] = b;
    }
    __syncthreads();

    int wave = tid >> 5;                        // 0..7  -> output-channel tile
    int lane = tid & 31;
    int m    = lane & 15;                       // A row (M)
    int kOff = (lane >> 4) * 2;                 // K sub-offset within 4-step
    int nCol = wave * 16 + (lane & 15);         // output channel (N)

    const float* B = kern + k * (C_IN * C_OUT); // [64][128] row-major

    v8f acc = {};
#pragma unroll
    for (int kk = 0; kk < C_IN / 4; ++kk) {
        int kb = kk * 4 + kOff;
        v2f a, b;
        a.x = A[m][kb];
        a.y = A[m][kb + 1];
        b.x = B[kb * C_OUT + nCol];
        b.y = B[(kb + 1) * C_OUT + nCol];
        acc = __builtin_amdgcn_wmma_f32_16x16x4_f32(
                  false, a, false, b, (short)0, acc, false, false);
    }

    // ---- scatter C: lane<16 -> M=r, lane>=16 -> M=8+r ----
    int mHi = (lane >> 4) * 8;
    int colOfs = nCol * HW;
#pragma unroll
    for (int r = 0; r < 8; ++r) {
        int bI = base[r + mHi];
        if (bI >= 0) atomicAdd(out + bI + colOfs, acc[r]);
    }
}

extern "C" void kernel_launch(void* const* d_in, const int* in_sizes, int n_in,
                              void* d_out, int out_size, void* d_ws, size_t ws_size,
                              hipStream_t stream) {
    const int*   coords    = (const int*)d_in[0];
    const float* feats     = (const float*)d_in[1];
    const float* kern      = (const float*)d_in[2];
    const int*   stridePtr = (const int*)d_in[3];
    float*       out       = (float*)d_out;

    int n        = in_sizes[0] / 4;             // coords is [N,4]
    int maxTiles = (n + 15) / 16 + NKER;

    int* ws       = (int*)d_ws;
    int* counts   = ws;
    int* offsets  = ws + 16;
    int* cursor   = ws + 32;
    int* numTiles = ws + 48;
    int* tiles    = ws + 64;
    int* perm     = ws + 64 + 3 * maxTiles;

    int nb = (n + 255) / 256;

    zero_meta_k<<<1, 64, 0, stream>>>(ws);
    count_k<<<nb, 256, 0, stream>>>(coords, stridePtr, counts, n);
    scan_build_k<<<1, 1, 0, stream>>>(ws);
    scatter_k<<<nb, 256, 0, stream>>>(coords, stridePtr, offsets, cursor, perm, n);

    int n4 = out_size / 4;
    zero_out_k<<<(n4 + 255) / 256, 256, 0, stream>>>((float4*)out, n4);

    gemm_scatter_k<<<maxTiles, 256, 0, stream>>>(feats, kern, coords, perm,
                                                 tiles, numTiles, stridePtr, out);
}